// Lidar_group_5214090297523
// MI455X (gfx1250) — compile-verified
//
#include <hip/hip_runtime.h>
#include <stdint.h>

// ---------------------------------------------------------------------------
// LiDAR point grouping (PointPillars-style scatter) for MI455X / gfx1250.
//
// Reference: data [8, 200000, 4] f32 in [0,1) -> out [8, 512, 512, 9*4] f32.
// Each nonzero point is binned into cell (int(x*512), int(y*512)) and takes
// the first free of 9 depth slots *in original point order*, storing
// {x*512 - xi, y*512 - yi, z, w}.
//
// This is pure data movement (~0.5 GB total traffic -> ~21us at 23.3 TB/s).
// No matrix math exists here, so instead of WMMA we use the CDNA5 async
// LDS->global store engine (ASYNCcnt) for the sentinel-pattern fill, a
// deterministic atomicMin top-9 insertion cascade (order-independent final
// state == reference's in-order slot assignment), and b128 vmem throughout.
// ---------------------------------------------------------------------------

#define LG_SHAPE   512
#define LG_DEPTH   9
#define LG_NCELLS  (LG_SHAPE * LG_SHAPE)      // 262144
#define LG_NPTS    200000
#define LG_NBATCH  8
#define LG_FEATS   4
#define LG_SENT    0x7FFFFFFF                  // INT_MAX sentinel (> any index)

// ---------------------------------------------------------------------------
// Pass 1: initialize every 16B slot group to {sentinel_bits, 0, 0, 0} using
// the CDNA5 async store engine. The pattern lives in LDS; each wave issues
// one global_store_async_from_lds_b128 per lane-group per iteration — no
// data VGPRs needed, tracked with ASYNCcnt (drained before s_endpgm).
// ---------------------------------------------------------------------------
__global__ void lg_init_async(float* __restrict__ out, unsigned ngroups) {
    __shared__ __align__(16) unsigned pat[4];
    if (threadIdx.x == 0) {
        pat[0] = LG_SENT;   // slot-index float carries INT_MAX bits
        pat[1] = 0u;
        pat[2] = 0u;
        pat[3] = 0u;
    }
    __syncthreads();

    // Flat shared pointer: low 32 bits are the wave-relative LDS byte offset.
    unsigned lds_off = (unsigned)(uintptr_t)(&pat[0]);

    unsigned g      = blockIdx.x * blockDim.x + threadIdx.x;
    unsigned stride = gridDim.x * blockDim.x;
    for (; g < ngroups; g += stride) {
        uint64_t addr = (uint64_t)(uintptr_t)out + (uint64_t)g * 16u;
        asm volatile("global_store_async_from_lds_b128 %0, %1, off"
                     :: "v"(addr), "v"(lds_off)
                     : "memory");
    }
    asm volatile("s_wait_asynccnt 0x0" ::: "memory");
}

// ---------------------------------------------------------------------------
// Pass 2: one thread per point. Deterministic concurrent top-9 insertion:
// atomicMin cascade on the slot-index ints (stride 16B). Final state is the
// 9 smallest point indices of each cell in ascending order — exactly the
// reference's "first free slot in original point order", independent of
// thread scheduling.
// ---------------------------------------------------------------------------
__global__ void lg_insert(const float4* __restrict__ pts,
                          float* __restrict__ out) {
    unsigned i = blockIdx.x * blockDim.x + threadIdx.x;
    if (i >= (unsigned)(LG_NBATCH * LG_NPTS)) return;

    unsigned b  = i / LG_NPTS;
    unsigned pi = i - b * LG_NPTS;        // index within batch == rank key

    float4 p = pts[i];                    // global_load_b128
    if (p.x == 0.0f && p.y == 0.0f && p.z == 0.0f && p.w == 0.0f) return;

    float xs = p.x * (float)LG_SHAPE;
    float ys = p.y * (float)LG_SHAPE;
    int xi = (int)xs;                     // truncation == floor for [0,512)
    int yi = (int)ys;
    xi = min(max(xi, 0), LG_SHAPE - 1);   // defensive clamp (no-OOB)
    yi = min(max(yi, 0), LG_SHAPE - 1);
    unsigned cell = (unsigned)xi * LG_SHAPE + (unsigned)yi;

    float* base = out + ((size_t)b * LG_NCELLS + cell) * (LG_DEPTH * LG_FEATS);
    __builtin_prefetch(base, 1, 3);       // global_prefetch_b8: warm the line

    int cur = (int)pi;
#pragma unroll
    for (int s = 0; s < LG_DEPTH; ++s) {
        int old = atomicMin((int*)(base + s * LG_FEATS), cur);
        if (old == LG_SENT) break;        // hit an empty slot: done
        cur = max(cur, old);              // displaced value moves down
    }
    // If cur falls off the end, point had rank >= 9 -> dropped (matches ref).
}

// ---------------------------------------------------------------------------
// Pass 3: one thread per (batch, cell, slot) group — 36 floats/cell over
// 4-float groups means each group IS one slot. Read the 4B index float;
// sentinel -> overwrite with 0.0 (rest already zero); valid -> gather the
// point and write the full 16B payload in a single b128 store.
// ---------------------------------------------------------------------------
__global__ void lg_fill(const float4* __restrict__ pts,
                        float4* __restrict__ outv,
                        unsigned ngroups) {
    unsigned g = blockIdx.x * blockDim.x + threadIdx.x;
    if (g >= ngroups) return;

    const float* outs = (const float*)outv;
    int idx = __float_as_int(outs[(size_t)g * 4]);   // 4B load of index float

    if (idx == LG_SENT) {
        ((float*)outv)[(size_t)g * 4] = 0.0f;        // scalar fix-up
        return;
    }

    unsigned b = g / (unsigned)(LG_NCELLS * LG_DEPTH);
    float4 p = pts[(size_t)b * LG_NPTS + (unsigned)idx];  // gather b128

    float xs = p.x * (float)LG_SHAPE;
    float ys = p.y * (float)LG_SHAPE;
    float fx = xs - (float)(int)xs;
    float fy = ys - (float)(int)ys;

    float4 v;
    v.x = fx; v.y = fy; v.z = p.z; v.w = p.w;
    outv[g] = v;                                     // global_store_b128
}

// ---------------------------------------------------------------------------
// Launch: out_size = 8*512*512*36 = 75,497,472 floats -> 18,874,368 groups.
// 256-thread blocks = 8 wave32s (fills a WGP twice over).
// ---------------------------------------------------------------------------
extern "C" void kernel_launch(void* const* d_in, const int* in_sizes, int n_in,
                              void* d_out, int out_size, void* d_ws, size_t ws_size,
                              hipStream_t stream) {
    (void)in_sizes; (void)n_in; (void)d_ws; (void)ws_size;

    const float* pts = (const float*)d_in[0];
    float*       out = (float*)d_out;

    unsigned ngroups = (unsigned)(out_size / 4);     // 16B slot groups

    // Pass 1: async pattern fill (grid-stride; ~9 async ops in flight / wave).
    lg_init_async<<<dim3(8192), dim3(256), 0, stream>>>(out, ngroups);

    // Pass 2: deterministic slot insertion, one thread per point.
    unsigned npts_total = (unsigned)(LG_NBATCH * LG_NPTS);
    lg_insert<<<dim3((npts_total + 255) / 256), dim3(256), 0, stream>>>(
        (const float4*)pts, out);

    // Pass 3: resolve slots -> final payloads.
    lg_fill<<<dim3((ngroups + 255) / 256), dim3(256), 0, stream>>>(
        (const float4*)pts, (float4*)out, ngroups);
}